// PointTransformer_7387343749851
// MI455X (gfx1250) — compile-verified
//
#include <hip/hip_runtime.h>
#include <hip/hip_bf16.h>
#include <math.h>

typedef __attribute__((ext_vector_type(2))) float v2f;
typedef __attribute__((ext_vector_type(8))) float v8f;

#define N_PTS   32768
#define PPB     4096          // points per batch (B=8)
#define IN_C    64
#define MID     128
#define HD      16
#define KNN     16
#define EPS     1e-5f
#define CNT     ((float)((size_t)N_PTS * KNN))

// ---- workspace layout (float units); total ~23.07M floats ~= 92.3 MB ----
#define OFF_Q    ((size_t)0)
#define OFF_K    (OFF_Q  + (size_t)N_PTS * MID)
#define OFF_V    (OFF_K  + (size_t)N_PTS * MID)
#define OFF_T    (OFF_V  + (size_t)N_PTS * MID)        // t, later overwritten by a=relu(BN1(t))
#define OFF_H1   (OFF_T  + (size_t)N_PTS * KNN * 3)
#define OFF_ST   (OFF_H1 + (size_t)N_PTS * KNN * HD)   // stats block (512 floats)
#define OFF_IDX  (OFF_ST + 512)                        // int region, N*KNN ints
// stats indices: [0..2]=tSum [4..6]=tSq  [8..135]=attnSum [136..263]=attnSq
//                [264..279]=h1Sum [280..295]=h1Sq

// ---------------------------------------------------------------------------
// 1) QKV GEMM: [32768x64] x [64x128] (x3) with V_WMMA_F32_16X16X4_F32.
//    One wave -> one 16x16 output tile, K=64 in 16 wmma steps. fp32 exact.
// ---------------------------------------------------------------------------
__global__ void k_qkv(const float* __restrict__ x,
                      const float* __restrict__ wq, const float* __restrict__ bq,
                      const float* __restrict__ wk, const float* __restrict__ bk,
                      const float* __restrict__ wv, const float* __restrict__ bv,
                      float* __restrict__ q, float* __restrict__ k, float* __restrict__ v)
{
    const int wave = threadIdx.x >> 5;
    const int lane = threadIdx.x & 31;
    const int tile = blockIdx.x * 4 + wave;        // (N/16)*24 tiles total
    const int mt   = tile / 24;
    const int ct   = tile % 24;
    const int mat  = ct >> 3;                      // 0=q 1=k 2=v
    const int n0   = (ct & 7) * 16;

    const float* W  = (mat == 0) ? wq : (mat == 1) ? wk : wv;
    const float* bb = (mat == 0) ? bq : (mat == 1) ? bk : bv;
    float*       O  = (mat == 0) ? q  : (mat == 1) ? k  : v;

    const int row   = (mt << 4) + (lane & 15);     // A: M = lane&15
    const int col   = n0 + (lane & 15);            // B/D: N = lane&15
    const int khalf = (lane >> 4) << 1;            // lanes 16..31 hold K=2,3

    v8f acc = {};
    #pragma unroll
    for (int kc = 0; kc < IN_C; kc += 4) {
        v2f a, b;
        a.x = x[(size_t)row * IN_C + kc + khalf];
        a.y = x[(size_t)row * IN_C + kc + khalf + 1];
        b.x = W[(size_t)(kc + khalf)     * MID + col];
        b.y = W[(size_t)(kc + khalf + 1) * MID + col];
        acc = __builtin_amdgcn_wmma_f32_16x16x4_f32(
                  /*neg_a=*/false, a, /*neg_b=*/false, b,
                  /*c_mod=*/(short)0, acc, /*reuse_a=*/false, /*reuse_b=*/false);
    }
    const float bias  = bb[col];
    const int   mbase = (mt << 4) + ((lane >> 4) << 3);   // D: rows r / r+8 split
    #pragma unroll
    for (int r = 0; r < 8; ++r)
        O[(size_t)(mbase + r) * MID + col] = acc[r] + bias;
}

// ---------------------------------------------------------------------------
// 2) Batched KNN: one thread per query point, LDS-tiled candidates,
//    register top-16 (ascending d2 == jax top_k(-d2) order incl. self).
// ---------------------------------------------------------------------------
__global__ void k_knn(const float* __restrict__ pos, int* __restrict__ idx)
{
    __shared__ float sp[3 * 128];
    const int gid  = blockIdx.x * 128 + threadIdx.x;
    const int base = gid & ~(PPB - 1);             // batch start (PPB pow2)
    const float px = pos[(size_t)gid * 3 + 0];
    const float py = pos[(size_t)gid * 3 + 1];
    const float pz = pos[(size_t)gid * 3 + 2];

    float bd[KNN]; int bi[KNN];
    #pragma unroll
    for (int i = 0; i < KNN; ++i) { bd[i] = 3.4e38f; bi[i] = 0; }

    for (int j0 = 0; j0 < PPB; j0 += 128) {
        __syncthreads();
        #pragma unroll
        for (int c = 0; c < 3; ++c)
            sp[c * 128 + threadIdx.x] = pos[(size_t)(base + j0 + threadIdx.x) * 3 + c];
        __syncthreads();
        if (j0 + 128 < PPB)   // gfx1250 global_prefetch_b8 for the next tile
            __builtin_prefetch(&pos[(size_t)(base + j0 + 128 + threadIdx.x) * 3], 0, 1);

        for (int j = 0; j < 128; ++j) {
            const float dx = px - sp[j];
            const float dy = py - sp[128 + j];
            const float dz = pz - sp[256 + j];
            float d2 = dx * dx + dy * dy + dz * dz;
            if (d2 < bd[KNN - 1]) {
                bd[KNN - 1] = d2; bi[KNN - 1] = j0 + j;
                #pragma unroll
                for (int t = KNN - 1; t > 0; --t) {   // one bubble pass (list was sorted)
                    if (bd[t] < bd[t - 1]) {
                        float td = bd[t]; bd[t] = bd[t - 1]; bd[t - 1] = td;
                        int   ti = bi[t]; bi[t] = bi[t - 1]; bi[t - 1] = ti;
                    }
                }
            }
        }
    }
    #pragma unroll
    for (int i = 0; i < KNN; ++i) idx[(size_t)gid * KNN + i] = bi[i];   // within-batch index
}

// ---------------------------------------------------------------------------
// 3) zero the stats block
// ---------------------------------------------------------------------------
__global__ void k_zero(float* __restrict__ st)
{
    if (threadIdx.x < 512) st[threadIdx.x] = 0.f;
}

// ---------------------------------------------------------------------------
// 4) t = rel@wp1 + bp1 ; accumulate per-channel sum/sumsq (3 channels)
//    wave32 shuffle reduce -> LDS across 8 waves -> 6 global float atomics.
// ---------------------------------------------------------------------------
__global__ void k_tstats(const float* __restrict__ pos, const int* __restrict__ idx,
                         const float* __restrict__ wp1, const float* __restrict__ bp1,
                         float* __restrict__ t, float* __restrict__ st)
{
    __shared__ float red[8][6];
    const int gid  = blockIdx.x * 256 + threadIdx.x;    // pair id (n*16+s)
    const int n    = gid >> 4;
    const int base = n & ~(PPB - 1);
    const int j    = idx[gid];
    const float rx = pos[(size_t)(base + j) * 3 + 0] - pos[(size_t)n * 3 + 0];
    const float ry = pos[(size_t)(base + j) * 3 + 1] - pos[(size_t)n * 3 + 1];
    const float rz = pos[(size_t)(base + j) * 3 + 2] - pos[(size_t)n * 3 + 2];

    float tv[3];
    #pragma unroll
    for (int c = 0; c < 3; ++c) {
        tv[c] = rx * wp1[0 * 3 + c] + ry * wp1[1 * 3 + c] + rz * wp1[2 * 3 + c] + bp1[c];
        t[(size_t)gid * 3 + c] = tv[c];
    }
    float acc[6] = { tv[0], tv[1], tv[2], tv[0]*tv[0], tv[1]*tv[1], tv[2]*tv[2] };
    #pragma unroll
    for (int u = 0; u < 6; ++u)
        for (int off = 16; off > 0; off >>= 1)
            acc[u] += __shfl_xor(acc[u], off, 32);
    const int wave = threadIdx.x >> 5, lane = threadIdx.x & 31;
    if (lane == 0)
        for (int u = 0; u < 6; ++u) red[wave][u] = acc[u];
    __syncthreads();
    if (threadIdx.x < 6) {
        float s = 0.f;
        for (int w = 0; w < 8; ++w) s += red[w][threadIdx.x];
        const int u = threadIdx.x;                         // 0..2 -> sum@0, 3..5 -> sq@4
        atomicAdd(&st[(u < 3 ? 0 : 4) + (u % 3)], s);
    }
}

// ---------------------------------------------------------------------------
// 5) a = relu(BN1(t)) (overwrites t); pe = a@wp2+bp2; attn = k_g - q + pe;
//    accumulate attn sum/sumsq per channel (128 channels == threadIdx.x).
// ---------------------------------------------------------------------------
__global__ void k_attn(const float* __restrict__ q, const float* __restrict__ kk,
                       const int* __restrict__ idx, float* __restrict__ t,
                       const float* __restrict__ gp, const float* __restrict__ betap,
                       const float* __restrict__ wp2, const float* __restrict__ bp2,
                       float* __restrict__ st)
{
    const int c = threadIdx.x;                       // channel 0..127
    float m1[3], i1[3], gg[3], bb[3];
    #pragma unroll
    for (int u = 0; u < 3; ++u) {
        m1[u] = st[u] / CNT;
        const float var = st[4 + u] / CNT - m1[u] * m1[u];
        i1[u] = rsqrtf(var + EPS);
        gg[u] = gp[u]; bb[u] = betap[u];
    }
    const float wc0 = wp2[0 * MID + c], wc1 = wp2[1 * MID + c], wc2 = wp2[2 * MID + c];
    const float bpc = bp2[c];
    float ls = 0.f, lq = 0.f;
    const int p0 = blockIdx.x * 64;
    for (int g = 0; g < 64; ++g) {
        const int pair = p0 + g;
        const int n    = pair >> 4;
        const int base = n & ~(PPB - 1);
        const int j    = idx[pair];
        float a0 = t[(size_t)pair * 3 + 0];
        float a1 = t[(size_t)pair * 3 + 1];
        float a2 = t[(size_t)pair * 3 + 2];
        a0 = fmaxf((a0 - m1[0]) * i1[0] * gg[0] + bb[0], 0.f);
        a1 = fmaxf((a1 - m1[1]) * i1[1] * gg[1] + bb[1], 0.f);
        a2 = fmaxf((a2 - m1[2]) * i1[2] * gg[2] + bb[2], 0.f);
        __syncthreads();                             // all reads of t[pair] done
        if (c < 3) t[(size_t)pair * 3 + c] = (c == 0) ? a0 : (c == 1) ? a1 : a2;
        const float pe   = a0 * wc0 + a1 * wc1 + a2 * wc2 + bpc;
        const float attn = kk[(size_t)(base + j) * MID + c] - q[(size_t)n * MID + c] + pe;
        ls += attn; lq += attn * attn;
    }
    atomicAdd(&st[8 + c],   ls);
    atomicAdd(&st[136 + c], lq);
}

// ---------------------------------------------------------------------------
// 6) h1 = relu(BN2(attn)) @ w1 + bb1  (128 -> 16 via LDS segment partials),
//    store h1, accumulate h1 sum/sumsq (16 channels).
// ---------------------------------------------------------------------------
__global__ void k_h1(const float* __restrict__ q, const float* __restrict__ kk,
                     const int* __restrict__ idx, const float* __restrict__ t /* = a */,
                     const float* __restrict__ wp2, const float* __restrict__ bp2,
                     const float* __restrict__ g1, const float* __restrict__ b1,
                     const float* __restrict__ w1, const float* __restrict__ bb1,
                     float* __restrict__ h1, float* __restrict__ st)
{
    __shared__ float r[MID];
    __shared__ float part[8][HD];
    const int c  = threadIdx.x;
    const float m2  = st[8 + c] / CNT;
    const float vr2 = st[136 + c] / CNT - m2 * m2;
    const float i2  = rsqrtf(vr2 + EPS);
    const float g1c = g1[c], b1c = b1[c];
    const float wc0 = wp2[c], wc1 = wp2[MID + c], wc2 = wp2[2 * MID + c], bpc = bp2[c];
    const int i16 = c & 15, seg = c >> 4;
    float ls = 0.f, lq = 0.f;

    for (int g = 0; g < 32; ++g) {
        const int pair = blockIdx.x * 32 + g;
        const int n    = pair >> 4;
        const int base = n & ~(PPB - 1);
        const int j    = idx[pair];
        const float a0 = t[(size_t)pair * 3 + 0];
        const float a1 = t[(size_t)pair * 3 + 1];
        const float a2 = t[(size_t)pair * 3 + 2];
        const float pe   = a0 * wc0 + a1 * wc1 + a2 * wc2 + bpc;
        const float attn = kk[(size_t)(base + j) * MID + c] - q[(size_t)n * MID + c] + pe;
        __syncthreads();                       // previous iteration done with r/part
        r[c] = fmaxf((attn - m2) * i2 * g1c + b1c, 0.f);
        __syncthreads();
        float p = 0.f;
        #pragma unroll
        for (int u = 0; u < 16; ++u)
            p += r[seg * 16 + u] * w1[(size_t)(seg * 16 + u) * HD + i16];
        part[seg][i16] = p;
        __syncthreads();
        if (seg == 0) {
            float h = bb1[i16];
            #pragma unroll
            for (int s = 0; s < 8; ++s) h += part[s][i16];
            h1[(size_t)pair * HD + i16] = h;
            ls += h; lq += h * h;
        }
    }
    if (seg == 0) {
        atomicAdd(&st[264 + i16], ls);
        atomicAdd(&st[280 + i16], lq);
    }
}

// ---------------------------------------------------------------------------
// 7) per point: vpe = v_g + pe (LDS), h2 = relu(BN3(h1))@w2+bb2,
//    softmax over neighbors per hd-channel, weighted reduce -> out[n,128].
// ---------------------------------------------------------------------------
__global__ void k_final(const float* __restrict__ v, const int* __restrict__ idx,
                        const float* __restrict__ t /* = a */, const float* __restrict__ wp2,
                        const float* __restrict__ bp2, const float* __restrict__ h1,
                        const float* __restrict__ g2, const float* __restrict__ b2,
                        const float* __restrict__ w2, const float* __restrict__ bb2,
                        const float* __restrict__ st, float* __restrict__ out)
{
    __shared__ float vpe[KNN][MID];   // 8 KB
    __shared__ float hw[KNN][HD];     // logits -> softmax weights
    const int n    = blockIdx.x;
    const int base = n & ~(PPB - 1);
    const int tx   = threadIdx.x;

    // vpe = v_g + pe  (2048 elements, 8 per thread)
    #pragma unroll
    for (int rr = 0; rr < 8; ++rr) {
        const int e  = tx + 256 * rr;
        const int tt = e >> 7, c = e & 127;
        const int pair = n * KNN + tt;
        const int j    = idx[pair];
        const float a0 = t[(size_t)pair * 3 + 0];
        const float a1 = t[(size_t)pair * 3 + 1];
        const float a2 = t[(size_t)pair * 3 + 2];
        const float pe = a0 * wp2[c] + a1 * wp2[MID + c] + a2 * wp2[2 * MID + c] + bp2[c];
        vpe[tt][c] = v[(size_t)(base + j) * MID + c] + pe;
    }
    // h2 logits (16 neighbors x 16 channels)
    {
        const int tt = tx >> 4, i = tx & 15;
        float h = bb2[i];
        #pragma unroll
        for (int cc = 0; cc < HD; ++cc) {
            const float m3  = st[264 + cc] / CNT;
            const float vr3 = st[280 + cc] / CNT - m3 * m3;
            const float i3  = rsqrtf(vr3 + EPS);
            const float rv  = fmaxf((h1[(size_t)(n * KNN + tt) * HD + cc] - m3) * i3 * g2[cc] + b2[cc], 0.f);
            h += rv * w2[(size_t)cc * HD + i];
        }
        hw[tt][i] = h;
    }
    __syncthreads();
    // softmax over neighbors per channel i
    if (tx < HD) {
        const int i = tx;
        float mx = -3.4e38f;
        #pragma unroll
        for (int tt = 0; tt < KNN; ++tt) mx = fmaxf(mx, hw[tt][i]);
        float e[KNN]; float sum = 0.f;
        #pragma unroll
        for (int tt = 0; tt < KNN; ++tt) { e[tt] = __expf(hw[tt][i] - mx); sum += e[tt]; }
        const float inv = 1.f / sum;
        #pragma unroll
        for (int tt = 0; tt < KNN; ++tt) hw[tt][i] = e[tt] * inv;
    }
    __syncthreads();
    // out[n, s*16+i] = sum_t vpe[t][s*16+i] * w[t][i]
    if (tx < MID) {
        const int s = tx >> 4, i = tx & 15;
        float o = 0.f;
        #pragma unroll
        for (int tt = 0; tt < KNN; ++tt)
            o += vpe[tt][s * HD + i] * hw[tt][i];
        out[(size_t)n * MID + tx] = o;
    }
}

// ---------------------------------------------------------------------------
extern "C" void kernel_launch(void* const* d_in, const int* in_sizes, int n_in,
                              void* d_out, int out_size, void* d_ws, size_t ws_size,
                              hipStream_t stream)
{
    const float* pos   = (const float*)d_in[0];
    const float* x     = (const float*)d_in[1];
    // d_in[2] = offset (int64) unused: uniform batches, B=8, P=4096
    const float* wq    = (const float*)d_in[3];
    const float* bq    = (const float*)d_in[4];
    const float* wk    = (const float*)d_in[5];
    const float* bk    = (const float*)d_in[6];
    const float* wv    = (const float*)d_in[7];
    const float* bv    = (const float*)d_in[8];
    const float* wp1   = (const float*)d_in[9];
    const float* bp1   = (const float*)d_in[10];
    const float* gp    = (const float*)d_in[11];
    const float* betap = (const float*)d_in[12];
    const float* wp2   = (const float*)d_in[13];
    const float* bp2   = (const float*)d_in[14];
    const float* g1    = (const float*)d_in[15];
    const float* b1    = (const float*)d_in[16];
    const float* w1    = (const float*)d_in[17];
    const float* bb1   = (const float*)d_in[18];
    const float* g2    = (const float*)d_in[19];
    const float* b2    = (const float*)d_in[20];
    const float* w2    = (const float*)d_in[21];
    const float* bb2   = (const float*)d_in[22];
    float* out = (float*)d_out;

    float* ws  = (float*)d_ws;                // need ws_size >= ~92.3 MB
    float* q   = ws + OFF_Q;
    float* k   = ws + OFF_K;
    float* v   = ws + OFF_V;
    float* t   = ws + OFF_T;
    float* h1  = ws + OFF_H1;
    float* st  = ws + OFF_ST;
    int*   idx = (int*)(ws + OFF_IDX);

    k_qkv  <<<(N_PTS / 16) * 24 / 4, 128, 0, stream>>>(x, wq, bq, wk, bk, wv, bv, q, k, v);
    k_knn  <<<N_PTS / 128,           128, 0, stream>>>(pos, idx);
    k_zero <<<1,                     512, 0, stream>>>(st);
    k_tstats<<<N_PTS * KNN / 256,    256, 0, stream>>>(pos, idx, wp1, bp1, t, st);
    k_attn <<<N_PTS * KNN / 64,      128, 0, stream>>>(q, k, idx, t, gp, betap, wp2, bp2, st);
    k_h1   <<<N_PTS * KNN / 32,      128, 0, stream>>>(q, k, idx, t, wp2, bp2, g1, b1, w1, bb1, h1, st);
    k_final<<<N_PTS,                 256, 0, stream>>>(v, idx, t, wp2, bp2, h1, g2, b2, w2, bb2, st, out);
}